// GraphNetworkOptimizer_36086315221040
// MI455X (gfx1250) — compile-verified
//
#include <hip/hip_runtime.h>

typedef float v2f __attribute__((ext_vector_type(2)));
typedef float v8f __attribute__((ext_vector_type(8)));

// ---------------- degree / norm precompute (shared across 3 layers) ---------
__global__ void deg_init_kernel(float* __restrict__ deg, int n) {
    int i = blockIdx.x * blockDim.x + threadIdx.x;
    if (i < n) deg[i] = 1.0f;  // self-loop contributes 1
}

__global__ void deg_accum_kernel(const int* __restrict__ dst, float* __restrict__ deg, int e) {
    int i = blockIdx.x * blockDim.x + threadIdx.x;
    if (i < e) atomicAdd(&deg[dst[i]], 1.0f);
}

__global__ void deg_rsqrt_kernel(float* __restrict__ deg, int n) {
    int i = blockIdx.x * blockDim.x + threadIdx.x;
    if (i < n) deg[i] = rsqrtf(deg[i]);  // deg >= 1 always (self-loops)
}

// ---------------- fp32 WMMA GEMM: H[M x NOUT] = (relu?)X[M x 64] @ W[64 x NOUT]
// One wave per 16-row tile; NOUT/16 accumulators; K stepped by 4 via
// V_WMMA_F32_16X16X4_F32 (exact fp32 precision, wave32 layout per ISA 7.12.2).
template<int NOUT, bool RELU>
__global__ void gemm_wmma_kernel(const float* __restrict__ X,
                                 const float* __restrict__ W,
                                 float* __restrict__ H, int numTiles) {
    const int lane = threadIdx.x & 31;
    const int wave = threadIdx.x >> 5;
    const int tile = blockIdx.x * (blockDim.x >> 5) + wave;
    if (tile >= numTiles) return;              // wave-uniform: EXEC stays all-1s
    const int half = lane >> 4;                // 0: K=0,1 ; 1: K=2,3
    const int l15  = lane & 15;
    constexpr int NT = NOUT / 16;

    v8f acc[NT] = {};
    const float* xrow = X + (size_t)(tile * 16 + l15) * 64 + 2 * half;
#pragma unroll
    for (int k = 0; k < 64; k += 4) {
        v2f a = *(const v2f*)(xrow + k);       // A[M=l15][k+2*half, +1]
        if (RELU) { a.x = fmaxf(a.x, 0.0f); a.y = fmaxf(a.y, 0.0f); }
#pragma unroll
        for (int t = 0; t < NT; ++t) {
            const float* wr = W + (size_t)(k + 2 * half) * NOUT + t * 16 + l15;
            v2f b; b.x = wr[0]; b.y = wr[NOUT]; // B[k+2*half][n], B[k+2*half+1][n]
            acc[t] = __builtin_amdgcn_wmma_f32_16x16x4_f32(
                false, a, false, b, (short)0, acc[t], false, false);
        }
    }
    // C/D layout: lane n = l15; VGPR j holds row M = j + 8*half
#pragma unroll
    for (int t = 0; t < NT; ++t)
#pragma unroll
        for (int j = 0; j < 8; ++j)
            H[(size_t)(tile * 16 + j + 8 * half) * NOUT + t * 16 + l15] = acc[t][j];
}

// ---------------- out[i] = h[i]*dinv[i]^2 + bias  (self-loop term + bias) ----
template<int F>
__global__ void init_out_kernel(const float* __restrict__ h,
                                const float* __restrict__ dinv,
                                const float* __restrict__ bias,
                                float* __restrict__ out, int n) {
    constexpr int FG = F / 4;
    int gid = blockIdx.x * blockDim.x + threadIdx.x;
    if (gid >= n * FG) return;
    int i  = gid / FG;
    int f4 = gid % FG;
    float di = dinv[i];
    float s  = di * di;
    float4 v = ((const float4*)h)[gid];
    float4 b = ((const float4*)bias)[f4];
    float4 o;
    o.x = v.x * s + b.x; o.y = v.y * s + b.y;
    o.z = v.z * s + b.z; o.w = v.w * s + b.w;
    ((float4*)out)[gid] = o;
}

// ---------------- edge scatter: out[dst] += h[src] * dinv[src]*dinv[dst] -----
// F/4 lanes per edge, float4 gather (L2-resident), f32 atomic-add scatter.
template<int F>
__global__ void edge_scatter_kernel(const float* __restrict__ h,
                                    const int* __restrict__ src,
                                    const int* __restrict__ dst,
                                    const float* __restrict__ dinv,
                                    float* out, int e) {
    constexpr int FG = F / 4;
    int gid = blockIdx.x * blockDim.x + threadIdx.x;
    int ed  = gid / FG;
    if (ed >= e) return;
    int f4 = gid % FG;
    int s = src[ed];
    int d = dst[ed];
    float nrm = dinv[s] * dinv[d];
    float4 v = ((const float4*)h)[(size_t)s * FG + f4];
    float* o = out + (size_t)d * F + f4 * 4;
    atomicAdd(o + 0, v.x * nrm);
    atomicAdd(o + 1, v.y * nrm);
    atomicAdd(o + 2, v.z * nrm);
    atomicAdd(o + 3, v.w * nrm);
}

extern "C" void kernel_launch(void* const* d_in, const int* in_sizes, int n_in,
                              void* d_out, int out_size, void* d_ws, size_t ws_size,
                              hipStream_t stream) {
    const float* x  = (const float*)d_in[0];
    const int*   ei = (const int*)d_in[1];
    const float* W1 = (const float*)d_in[2];
    const float* b1 = (const float*)d_in[3];
    const float* W2 = (const float*)d_in[4];
    const float* b2 = (const float*)d_in[5];
    const float* W3 = (const float*)d_in[6];
    const float* b3 = (const float*)d_in[7];

    const int n = in_sizes[0] / 64;   // 100000
    const int e = in_sizes[1] / 2;    // 1600000
    const int* src = ei;
    const int* dst = ei + e;

    float* dinv = (float*)d_ws;                 // n floats
    float* buf0 = dinv + n;                     // n*64 floats (h / h3)
    float* buf1 = buf0 + (size_t)n * 64;        // n*64 floats (aggregated)
    float* out  = (float*)d_out;                // n*32 floats

    const int T = 256;

    // --- normalization factors (shared across layers) ---
    deg_init_kernel<<<(n + T - 1) / T, T, 0, stream>>>(dinv, n);
    deg_accum_kernel<<<(e + T - 1) / T, T, 0, stream>>>(dst, dinv, e);
    deg_rsqrt_kernel<<<(n + T - 1) / T, T, 0, stream>>>(dinv, n);

    const int tiles = (n + 15) / 16;            // 6250 (exact)
    const int waves = T / 32;
    dim3 gg((tiles + waves - 1) / waves);

    // --- layer 1: h = x@W1 ; out1 = scatter(h) + b1   (relu deferred) ---
    gemm_wmma_kernel<64, false><<<gg, T, 0, stream>>>(x, W1, buf0, tiles);
    init_out_kernel<64><<<(n * 16 + T - 1) / T, T, 0, stream>>>(buf0, dinv, b1, buf1, n);
    edge_scatter_kernel<64><<<(e * 16 + T - 1) / T, T, 0, stream>>>(buf0, src, dst, dinv, buf1, e);

    // --- layer 2: h = relu(out1)@W2 ; out2 = scatter(h) + b2 ---
    gemm_wmma_kernel<64, true><<<gg, T, 0, stream>>>(buf1, W2, buf0, tiles);
    init_out_kernel<64><<<(n * 16 + T - 1) / T, T, 0, stream>>>(buf0, dinv, b2, buf1, n);
    edge_scatter_kernel<64><<<(e * 16 + T - 1) / T, T, 0, stream>>>(buf0, src, dst, dinv, buf1, e);

    // --- layer 3: h = relu(out2)@W3 ; d_out = scatter(h) + b3 (no relu) ---
    gemm_wmma_kernel<32, true><<<gg, T, 0, stream>>>(buf1, W3, buf0, tiles);
    init_out_kernel<32><<<(n * 8 + T - 1) / T, T, 0, stream>>>(buf0, dinv, b3, out, n);
    edge_scatter_kernel<32><<<(e * 8 + T - 1) / T, T, 0, stream>>>(buf0, src, dst, dinv, out, e);
}